// CenterLoss_55001351193088
// MI455X (gfx1250) — compile-verified
//
#include <hip/hip_runtime.h>
#include <stdint.h>

typedef __attribute__((ext_vector_type(2))) float v2f;
typedef __attribute__((ext_vector_type(8))) float v8f;

#define NUM_FEAS   256
#define LR         0.5f
#define KSLICES    10      // split the 10000-wide column scan into 10 slices per row-block

__device__ __forceinline__ v2f ld_nt_f2(const float* p) {
    unsigned long long raw = __builtin_nontemporal_load((const unsigned long long*)p);
    union { unsigned long long u; v2f f; } cvt; cvt.u = raw;
    return cvt.f;
}

// ---------------- kernel 0: zero the scratch accumulators ----------------
__global__ void zero_ws(float* idxf, int* count, int n_idx, int n_cnt) {
    int t = blockIdx.x * blockDim.x + threadIdx.x;
    if (t < n_idx) idxf[t] = 0.0f;
    if (t < n_cnt) count[t] = 0;
}

// ---------------- kernel 1: streaming label scan with WMMA ----------------
// idx_partial[m] += sum_j labels[m][j] * j  over this wave's column slice,
// computed as D = A(16x4 labels) x B(4x16 broadcast col indices) + C.
// Exact in f32: labels entries are 0/1, indices < 2^24.
__global__ void __launch_bounds__(256)
label_index_wmma(const float* __restrict__ labels,
                 float* __restrict__ idxf,
                 int nRows, int nCols) {
    const int lane     = threadIdx.x & 31;
    const int wave     = threadIdx.x >> 5;
    const int rowBlock = blockIdx.x * (blockDim.x >> 5) + wave;  // 16-row block
    const int rowBase  = rowBlock * 16;
    if (rowBase >= nRows) return;                 // uniform per wave

    const int sliceW  = ((nCols + KSLICES * 4 - 1) / (KSLICES * 4)) * 4;
    const int colBase = blockIdx.y * sliceW;
    const int colEnd  = (colBase + sliceW < nCols) ? (colBase + sliceW) : nCols;
    if (colBase >= colEnd) return;                // uniform per wave

    // Chunks of 4 columns whose loads (col .. col+3) are fully in-bounds.
    // For nCols % 4 == 0 (C = 10000) this is every chunk -> branch-free hot loop.
    const int cols       = colEnd - colBase;
    const int fullChunks = cols >> 2;             // each covers 4 columns
    const int rem        = cols & 3;

    // A-layout (32-bit A 16x4): lanes 0-15 -> K=0 (V0), K=1 (V1); lanes 16-31 -> K=2,3
    const int m    = lane & 15;
    const int kOff = (lane >> 4) * 2;
    const float* rowPtr = labels + (size_t)(rowBase + m) * (size_t)nCols;
    const float* p      = rowPtr + colBase + kOff;

    v8f acc0 = {0.f, 0.f, 0.f, 0.f, 0.f, 0.f, 0.f, 0.f};
    v8f acc1 = acc0;
    float bx = (float)(colBase + kOff);

    int c = 0;
    // ---- branch-free, 4x unrolled: 4 NT b64 loads in flight per iteration ----
    for (; c + 4 <= fullChunks; c += 4) {
        v2f a0 = ld_nt_f2(p);
        v2f a1 = ld_nt_f2(p + 4);
        v2f a2 = ld_nt_f2(p + 8);
        v2f a3 = ld_nt_f2(p + 12);
        p += 16;
        v2f b0; b0.x = bx;         b0.y = bx + 1.0f;
        v2f b1; b1.x = bx +  4.0f; b1.y = bx +  5.0f;
        v2f b2; b2.x = bx +  8.0f; b2.y = bx +  9.0f;
        v2f b3; b3.x = bx + 12.0f; b3.y = bx + 13.0f;
        acc0 = __builtin_amdgcn_wmma_f32_16x16x4_f32(false, a0, false, b0, (short)0, acc0, false, false);
        acc1 = __builtin_amdgcn_wmma_f32_16x16x4_f32(false, a1, false, b1, (short)0, acc1, false, false);
        acc0 = __builtin_amdgcn_wmma_f32_16x16x4_f32(false, a2, false, b2, (short)0, acc0, false, false);
        acc1 = __builtin_amdgcn_wmma_f32_16x16x4_f32(false, a3, false, b3, (short)0, acc1, false, false);
        bx += 16.0f;
    }
    // ---- leftover full chunks ----
    for (; c < fullChunks; ++c) {
        v2f a = ld_nt_f2(p);
        p += 4;
        v2f b; b.x = bx; b.y = bx + 1.0f;
        acc0 = __builtin_amdgcn_wmma_f32_16x16x4_f32(false, a, false, b, (short)0, acc0, false, false);
        bx += 4.0f;
    }
    // ---- remainder columns (only when nCols % 4 != 0; never for C=10000) ----
    if (rem) {
        int col = colBase + fullChunks * 4;
        v2f a;
        a.x = (col + kOff     < colEnd) ? rowPtr[col + kOff]     : 0.0f;
        a.y = (col + kOff + 1 < colEnd) ? rowPtr[col + kOff + 1] : 0.0f;
        v2f b; b.x = bx; b.y = bx + 1.0f;
        acc0 = __builtin_amdgcn_wmma_f32_16x16x4_f32(false, a, false, b, (short)0, acc0, false, false);
    }

    // exact: each element of acc0/acc1 is a sum of distinct integer column ids
    v8f acc;
    #pragma unroll
    for (int j = 0; j < 8; ++j) acc[j] = acc0[j] + acc1[j];

    // D layout: lanes 0-15 N=lane, V0..7 = M 0..7; lanes 16-31 = M 8..15.
    // Every N column holds the same value; only N==0 lanes (0 and 16) commit.
    if ((lane & 15) == 0) {
        int mBase = rowBase + (lane >> 4) * 8;
        #pragma unroll
        for (int j = 0; j < 8; ++j)
            atomicAdd(&idxf[mBase + j], acc[j]);   // exact integer partials
    }
}

// ---------------- kernel 2: finalize indices + class histogram ----------------
__global__ void finalize_idx(const float* __restrict__ idxf, int* __restrict__ idx,
                             int* __restrict__ count, int n, int C) {
    int i = blockIdx.x * blockDim.x + threadIdx.x;
    if (i >= n) return;
    int v = (int)(idxf[i] + 0.5f);
    v = v < 0 ? 0 : (v >= C ? C - 1 : v);
    idx[i] = v;
    atomicAdd(&count[v], 1);
}

// ---------------- kernel 3: out = center * (1 - LR*n_c) ----------------
__global__ void scale_center(const float4* __restrict__ center4,
                             const int* __restrict__ count,
                             float4* __restrict__ out4,
                             int totalVec, int vecPerRow) {
    int t = blockIdx.x * blockDim.x + threadIdx.x;
    if (t >= totalVec) return;
    int c = t / vecPerRow;
    float s = 1.0f - LR * (float)count[c];
    float4 v = center4[t];
    out4[t] = make_float4(v.x * s, v.y * s, v.z * s, v.w * s);
}

// ---------------- kernel 4: scatter-add LR*pred rows onto class rows ----------------
__global__ void scatter_preds(const float* __restrict__ preds,
                              const int* __restrict__ idx,
                              float* __restrict__ out, int total, int D) {
    int t = blockIdx.x * blockDim.x + threadIdx.x;
    if (t >= total) return;
    int i = t / D;
    int d = t - i * D;
    atomicAdd(&out[(size_t)idx[i] * D + d], LR * preds[t]);
}

extern "C" void kernel_launch(void* const* d_in, const int* in_sizes, int n_in,
                              void* d_out, int out_size, void* d_ws, size_t ws_size,
                              hipStream_t stream) {
    const float* preds  = (const float*)d_in[0];   // [N, 256] f32
    const float* labels = (const float*)d_in[1];   // [N, C]   f32 one-hot
    const float* center = (const float*)d_in[2];   // [C, 256] f32
    float*       out    = (float*)d_out;           // [C, 256] f32

    const int N = in_sizes[0] / NUM_FEAS;          // 8192
    const int C = in_sizes[2] / NUM_FEAS;          // 10000

    // workspace: [N floats idxf][N ints idx][C ints count]  (~105 KB)
    float* idxf  = (float*)d_ws;
    int*   idx   = (int*)((char*)d_ws + (size_t)N * sizeof(float));
    int*   count = idx + N;

    // 0) zero accumulators (ws is not re-poisoned between replays)
    int zn = (N > C) ? N : C;
    zero_ws<<<(zn + 255) / 256, 256, 0, stream>>>(idxf, count, N, C);

    // 1) WMMA streaming scan of labels (dominant 327.7 MB read)
    const int rowBlocks   = (N + 15) / 16;         // 512
    const int wavesPerBlk = 8;                     // 256-thread blocks = 8 waves
    dim3 g1((rowBlocks + wavesPerBlk - 1) / wavesPerBlk, KSLICES);
    label_index_wmma<<<g1, 32 * wavesPerBlk, 0, stream>>>(labels, idxf, N, C);

    // 2) finalize indices + per-class counts
    finalize_idx<<<(N + 255) / 256, 256, 0, stream>>>(idxf, idx, count, N, C);

    // 3) out = center * (1 - LR * n_c)
    const int vecPerRow = NUM_FEAS / 4;
    const int totalVec  = C * vecPerRow;
    scale_center<<<(totalVec + 255) / 256, 256, 0, stream>>>(
        (const float4*)center, count, (float4*)out, totalVec, vecPerRow);

    // 4) out[idx[i]] += LR * pred[i]
    const int total = N * NUM_FEAS;
    scatter_preds<<<(total + 255) / 256, 256, 0, stream>>>(preds, idx, out, total, NUM_FEAS);
}